// GCN_61744449848107
// MI455X (gfx1250) — compile-verified
//
#include <hip/hip_runtime.h>

typedef __attribute__((ext_vector_type(2))) float v2f;
typedef __attribute__((ext_vector_type(8))) float v8f;

#define N_NODES  100000   // divisible by 16
#define N_EDGES  3200000
#define N_GRAPHS 1024     // divisible by 16

static inline int ceil_div(int a, int b) { return (a + b - 1) / b; }

// ---------------- utility: vectorized fill (n must be a multiple of 4) ----------------
__global__ void fill4_f32(float4* __restrict__ p, float v, int nquads) {
    int i = blockIdx.x * blockDim.x + threadIdx.x;
    if (i < nquads) p[i] = make_float4(v, v, v, v);
}

// degree with self-loop already folded in via fill(1.0)
__global__ void deg_accum(const int* __restrict__ ei, float* __restrict__ deg) {
    int e = blockIdx.x * blockDim.x + threadIdx.x;
    if (e < N_EDGES) atomicAdd(&deg[ei[N_EDGES + e]], 1.0f);
}

__global__ void rsqrt_inplace(float* __restrict__ d, int n) {
    int i = blockIdx.x * blockDim.x + threadIdx.x;
    if (i < n) d[i] = rsqrtf(d[i]);
}

// ---------------- WMMA fp32 node transform: out = H @ W^T (+bias) ----------------
// H: [nrows, CIN] row-major, W: [COUT, CIN] row-major, out: [nrows, COUT].
// nrows MUST be a multiple of 16 (true for all call sites) -> no row guards, no
// divergence, EXEC all-ones at every V_WMMA_F32_16X16X4_F32.
// CIN not a multiple of 4 is handled with clamped in-bounds loads + value selects
// (VALU cndmask, no exec branches).
template <int CIN, int COUT>
__global__ void __launch_bounds__(256)
gemm_wmma_f32(const float* __restrict__ H, const float* __restrict__ W,
              const float* __restrict__ bias, float* __restrict__ out, int nrows) {
    constexpr int NT     = COUT / 16;        // 16-wide output tiles (1 or 2)
    constexpr int KSTEPS = (CIN + 3) / 4;    // K=4 per WMMA, zero-padded

    const int wavesPerBlock = blockDim.x >> 5;
    const int wave = blockIdx.x * wavesPerBlock + (threadIdx.x >> 5);
    const int row0 = wave * 16;
    if (row0 >= nrows) return;               // wave-uniform

    const int lane = threadIdx.x & 31;
    const int half = lane >> 4;              // 0: K pair {0,1}, 1: K pair {2,3}
    const int m    = lane & 15;
    const int row  = row0 + m;               // A-matrix row for this lane

    v8f acc[NT];
#pragma unroll
    for (int nt = 0; nt < NT; ++nt) acc[nt] = (v8f){};

#pragma unroll
    for (int ks = 0; ks < KSTEPS; ++ks) {
        const int ka = ks * 4 + 2 * half;    // first K of this lane's pair

        // A tile: 16x4 fp32 (ISA layout: lanes 0-15 = K0/K1, lanes 16-31 = K2/K3)
        v2f a;
        if constexpr (CIN % 4 == 0) {
            a = *reinterpret_cast<const v2f*>(&H[row * CIN + ka]);   // global_load_b64
        } else {
            const int k0 = (ka     < CIN) ? ka     : (CIN - 1);      // clamp in-bounds
            const int k1 = (ka + 1 < CIN) ? ka + 1 : (CIN - 1);
            float ax = H[row * CIN + k0];
            float ay = H[row * CIN + k1];
            a.x = (ka     < CIN) ? ax : 0.0f;                        // v_cndmask
            a.y = (ka + 1 < CIN) ? ay : 0.0f;
        }

#pragma unroll
        for (int nt = 0; nt < NT; ++nt) {
            const int ncol = nt * 16 + m;    // output channel for this lane's B column
            // B tile: 4x16 fp32, B[k][n] = W^T[k][n] = W[n][k]
            v2f b;
            if constexpr (CIN % 4 == 0) {
                b = *reinterpret_cast<const v2f*>(&W[ncol * CIN + ka]);
            } else {
                const int k0 = (ka     < CIN) ? ka     : (CIN - 1);
                const int k1 = (ka + 1 < CIN) ? ka + 1 : (CIN - 1);
                float bx = W[ncol * CIN + k0];
                float by = W[ncol * CIN + k1];
                b.x = (ka     < CIN) ? bx : 0.0f;
                b.y = (ka + 1 < CIN) ? by : 0.0f;
            }
            acc[nt] = __builtin_amdgcn_wmma_f32_16x16x4_f32(
                false, a, false, b, (short)0, acc[nt], false, false);
        }
    }

    // C/D layout: VGPR r -> row r (lanes 0-15) / row r+8 (lanes 16-31), col = lane&15.
    // For fixed r, lanes 0-15 write 64B contiguous, lanes 16-31 write 64B contiguous.
#pragma unroll
    for (int nt = 0; nt < NT; ++nt) {
        const int col = nt * 16 + m;
        const float bb = bias ? bias[col] : 0.0f;
#pragma unroll
        for (int r = 0; r < 8; ++r) {
            out[(row0 + r + 8 * half) * COUT + col] = acc[nt][r] + bb;
        }
    }
}

// ---------------- edge scatter: agg[dst] += t[src] * dis[src]*dis[dst] ----------------
// Each thread owns a 4-channel chunk of one edge: one coalesced b128 gather of the
// t[src] row segment + 4 f32 L2 atomics into agg[dst].
template <int COUT>
__global__ void edge_scatter(const int* __restrict__ ei, const float* __restrict__ t,
                             const float* __restrict__ dis, float* __restrict__ agg) {
    constexpr int GROUPS = COUT / 4;
    const int tid = blockIdx.x * blockDim.x + threadIdx.x;
    const int e = tid / GROUPS;
    const int g = tid % GROUPS;
    if (e >= N_EDGES) return;
    if (g == 0 && e + 4096 < N_EDGES) {      // stream the edge list ahead of use
        __builtin_prefetch(&ei[e + 4096], 0, 1);
        __builtin_prefetch(&ei[N_EDGES + e + 4096], 0, 1);
    }
    const int src = ei[e];
    const int dst = ei[N_EDGES + e];
    const float coeff = dis[src] * dis[dst];
    const float4 v = *reinterpret_cast<const float4*>(&t[src * COUT + g * 4]);
    float* base = &agg[dst * COUT + g * 4];
    atomicAdd(base + 0, v.x * coeff);
    atomicAdd(base + 1, v.y * coeff);
    atomicAdd(base + 2, v.z * coeff);
    atomicAdd(base + 3, v.w * coeff);
}

// ---------------- finalize: Hnext = relu(agg + t*dis^2 + b), float4 path ----------------
template <int COUT>
__global__ void finalize_relu(const float* __restrict__ agg, const float* __restrict__ t,
                              const float* __restrict__ dis, const float* __restrict__ bias,
                              float* __restrict__ outH) {
    constexpr int GROUPS = COUT / 4;
    const int q = blockIdx.x * blockDim.x + threadIdx.x;
    if (q >= N_NODES * GROUPS) return;
    const int n  = q / GROUPS;
    const int c4 = (q % GROUPS) * 4;
    const float d  = dis[n];
    const float dd = d * d;
    const float4 va = *reinterpret_cast<const float4*>(&agg[n * COUT + c4]);
    const float4 vt = *reinterpret_cast<const float4*>(&t[n * COUT + c4]);
    const float4 vb = *reinterpret_cast<const float4*>(&bias[c4]);
    float4 r;
    r.x = fmaxf(fmaf(vt.x, dd, va.x) + vb.x, 0.0f);
    r.y = fmaxf(fmaf(vt.y, dd, va.y) + vb.y, 0.0f);
    r.z = fmaxf(fmaf(vt.z, dd, va.z) + vb.z, 0.0f);
    r.w = fmaxf(fmaf(vt.w, dd, va.w) + vb.w, 0.0f);
    *reinterpret_cast<float4*>(&outH[n * COUT + c4]) = r;
}

// ---------------- global add pool: pooled[batch[n]] += h[n], 4 channels/thread ----------------
__global__ void pool_kernel(const float* __restrict__ h, const int* __restrict__ batch,
                            float* __restrict__ pooled) {
    const int q = blockIdx.x * blockDim.x + threadIdx.x;   // N_NODES * 4 quads
    if (q >= N_NODES * 4) return;
    const int n  = q >> 2;
    const int c4 = (q & 3) * 4;
    const float4 v = *reinterpret_cast<const float4*>(&h[n * 16 + c4]);
    float* base = &pooled[batch[n] * 16 + c4];
    atomicAdd(base + 0, v.x);
    atomicAdd(base + 1, v.y);
    atomicAdd(base + 2, v.z);
    atomicAdd(base + 3, v.w);
}

// ---------------- classifier: out[g] = relu(emb[g]) . Wc + bc ----------------
__global__ void classify(const float* __restrict__ emb, const float* __restrict__ Wc,
                         const float* __restrict__ bc, float* __restrict__ out) {
    const int g = blockIdx.x * blockDim.x + threadIdx.x;
    if (g >= N_GRAPHS) return;
    float s = bc[0];
#pragma unroll
    for (int c = 0; c < 16; ++c) s += fmaxf(emb[g * 16 + c], 0.0f) * Wc[c];
    out[g] = s;
}

extern "C" void kernel_launch(void* const* d_in, const int* in_sizes, int n_in,
                              void* d_out, int out_size, void* d_ws, size_t ws_size,
                              hipStream_t stream) {
    const float* x     = (const float*)d_in[0];
    const int*   ei    = (const int*)d_in[1];
    const int*   batch = (const int*)d_in[2];
    const float* W1 = (const float*)d_in[3];
    const float* b1 = (const float*)d_in[4];
    const float* W2 = (const float*)d_in[5];
    const float* b2 = (const float*)d_in[6];
    const float* W3 = (const float*)d_in[7];
    const float* b3 = (const float*)d_in[8];
    const float* We = (const float*)d_in[9];
    const float* be = (const float*)d_in[10];
    const float* Wc = (const float*)d_in[11];
    const float* bc = (const float*)d_in[12];
    float* out = (float*)d_out;              // [0,16384): embedding, [16384,17408): logits

    // workspace layout (floats; every buffer 16B-aligned)
    float* ws     = (float*)d_ws;
    float* dis    = ws;                       // N_NODES
    float* bufT   = dis  + N_NODES;           // N_NODES*32  (transform result)
    float* bufA   = bufT + N_NODES * 32;      // N_NODES*32  (aggregation)
    float* bufH   = bufA + N_NODES * 32;      // N_NODES*32  (layer features)
    float* pooled = bufH + N_NODES * 32;      // N_GRAPHS*16

    const int B = 256;

    // symmetric normalization: dis = rsqrt(deg + 1)
    fill4_f32<<<ceil_div(N_NODES / 4, B), B, 0, stream>>>((float4*)dis, 1.0f, N_NODES / 4);
    deg_accum<<<ceil_div(N_EDGES, B), B, 0, stream>>>(ei, dis);
    rsqrt_inplace<<<ceil_div(N_NODES, B), B, 0, stream>>>(dis, N_NODES);

    const int gBlocks = ceil_div(N_NODES / 16, B / 32);  // 8 waves / block

    // ---- layer 1: 3 -> 32 ----
    gemm_wmma_f32<3, 32><<<gBlocks, B, 0, stream>>>(x, W1, nullptr, bufT, N_NODES);
    fill4_f32<<<ceil_div(N_NODES * 8, B), B, 0, stream>>>((float4*)bufA, 0.0f, N_NODES * 8);
    edge_scatter<32><<<ceil_div(N_EDGES * 8, B), B, 0, stream>>>(ei, bufT, dis, bufA);
    finalize_relu<32><<<ceil_div(N_NODES * 8, B), B, 0, stream>>>(bufA, bufT, dis, b1, bufH);

    // ---- layer 2: 32 -> 16 ----
    gemm_wmma_f32<32, 16><<<gBlocks, B, 0, stream>>>(bufH, W2, nullptr, bufT, N_NODES);
    fill4_f32<<<ceil_div(N_NODES * 4, B), B, 0, stream>>>((float4*)bufA, 0.0f, N_NODES * 4);
    edge_scatter<16><<<ceil_div(N_EDGES * 4, B), B, 0, stream>>>(ei, bufT, dis, bufA);
    finalize_relu<16><<<ceil_div(N_NODES * 4, B), B, 0, stream>>>(bufA, bufT, dis, b2, bufH);

    // ---- layer 3: 16 -> 16 ----
    gemm_wmma_f32<16, 16><<<gBlocks, B, 0, stream>>>(bufH, W3, nullptr, bufT, N_NODES);
    fill4_f32<<<ceil_div(N_NODES * 4, B), B, 0, stream>>>((float4*)bufA, 0.0f, N_NODES * 4);
    edge_scatter<16><<<ceil_div(N_EDGES * 4, B), B, 0, stream>>>(ei, bufT, dis, bufA);
    finalize_relu<16><<<ceil_div(N_NODES * 4, B), B, 0, stream>>>(bufA, bufT, dis, b3, bufH);

    // ---- pooling + head ----
    fill4_f32<<<ceil_div(N_GRAPHS * 4, B), B, 0, stream>>>((float4*)pooled, 0.0f, N_GRAPHS * 4);
    pool_kernel<<<ceil_div(N_NODES * 4, B), B, 0, stream>>>(bufH, batch, pooled);
    gemm_wmma_f32<16, 16><<<ceil_div(N_GRAPHS / 16, B / 32), B, 0, stream>>>(
        pooled, We, be, out, N_GRAPHS);
    classify<<<ceil_div(N_GRAPHS, B), B, 0, stream>>>(out, Wc, bc, out + N_GRAPHS * 16);
}